// VectorQuantizer_5403068858626
// MI455X (gfx1250) — compile-verified
//
#include <hip/hip_runtime.h>
#include <math.h>
#include <stdint.h>

typedef __attribute__((ext_vector_type(2))) float v2f;
typedef __attribute__((ext_vector_type(8))) float v8f;

#define K_CB   1024          // codebook size
#define C_DIM  256           // embedding dim
#define HW     1024          // H*W
#define N_TOT  32768         // B*H*W
#define BCHW   8388608       // 32*256*32*32

// ---------------------------------------------------------------------------
// Kernel 1: per-code ||e||^2 and per-dim column sums of emb
// ---------------------------------------------------------------------------
__global__ __launch_bounds__(256) void vq_prep_emb(const float* __restrict__ emb,
                                                   float* __restrict__ enorm,
                                                   float* __restrict__ Ec) {
    int t = blockIdx.x * 256 + threadIdx.x;
    if (t < K_CB) {
        const float* e = emb + (size_t)t * C_DIM;
        float s = 0.f;
        for (int c = 0; c < C_DIM; ++c) s += e[c] * e[c];
        enorm[t] = s;
    } else if (t < K_CB + C_DIM) {
        int c = t - K_CB;
        float s = 0.f;
        for (int k = 0; k < K_CB; ++k) s += emb[(size_t)k * C_DIM + c];
        Ec[c] = s;
    }
}

// ---------------------------------------------------------------------------
// Kernel 2: per-dim column sums and sum-of-squares of z (NCHW layout)
// ---------------------------------------------------------------------------
__global__ __launch_bounds__(256) void vq_prep_z(const float* __restrict__ z,
                                                 float* __restrict__ Zc,
                                                 float* __restrict__ SqC) {
    __shared__ float s1[256];
    __shared__ float s2[256];
    const int c = blockIdx.x;
    const int tid = threadIdx.x;
    float a = 0.f, b = 0.f;
    for (int i = tid; i < N_TOT; i += 256) {
        float v = z[(size_t)(i >> 10) * (C_DIM * HW) + (size_t)c * HW + (i & (HW - 1))];
        a += v;
        b += v * v;
    }
    s1[tid] = a; s2[tid] = b;
    __syncthreads();
    for (int off = 128; off > 0; off >>= 1) {
        if (tid < off) { s1[tid] += s1[tid + off]; s2[tid] += s2[tid + off]; }
        __syncthreads();
    }
    if (tid == 0) { Zc[c] = s1[0]; SqC[c] = s2[0]; }
}

// ---------------------------------------------------------------------------
// Async staging of one emb tile (16 codes x 256 dims = 16 KB) into LDS.
// LDS layout: element (c, n) at float index (c>>2)*64 + n*4 + (c&3), so
//  - one GLOBAL_LOAD_ASYNC_TO_LDS_B128 moves 4 consecutive dims of one code
//    row to a contiguous 16B LDS slot (copy-compatible layout), and
//  - a WMMA B fragment (K = 2h+j) is one conflict-free ds_load_b64.
// Each of the 256 threads issues 4 async b128 copies (1024 total = 16 KB).
// ---------------------------------------------------------------------------
__device__ __forceinline__ void stage_tile_async(uint32_t lds_base_bytes,
                                                 unsigned long long emb_base,
                                                 int k0, int sn, int cb) {
#pragma unroll
    for (int i = 0; i < 4; ++i) {
        uint32_t ldsoff = lds_base_bytes +
                          (uint32_t)(((((cb >> 2) + i) * 64) + (sn << 2)) * 4);
        uint32_t goff = (uint32_t)(((((k0 + sn) << 8) + cb + (i << 2))) * 4);
        asm volatile("global_load_async_to_lds_b128 %0, %1, %2"
                     :: "v"(ldsoff), "v"(goff), "s"(emb_base)
                     : "memory");
    }
}

__device__ __forceinline__ void wait_async_zero() {
    asm volatile("s_wait_asynccnt 0x0" ::: "memory");
}

// ---------------------------------------------------------------------------
// Kernel 3 (main): WMMA distance GEMM + running argmin.
// 256 threads = 8 waves; wave w owns rows [blk*128 + w*16, +16).
// A panel (16 rows x 256 dims) held in 64 float2 fragments (128 VGPRs).
// K loop: 64 steps of 16 codes; emb tile double-buffered in LDS via
// async global->LDS copies (ASYNCcnt), one barrier per step.
// ---------------------------------------------------------------------------
__global__ __launch_bounds__(256) void vq_main(const float* __restrict__ z,
                                               const float* __restrict__ emb,
                                               const float* __restrict__ enorm,
                                               float* __restrict__ scores_out,
                                               float* __restrict__ idxf_out,
                                               int*   __restrict__ idx_ws,
                                               int*   __restrict__ counts,
                                               float* __restrict__ partials) {
    __shared__ float ebuf[2][4096];      // 2 x 16 KB tiles
    __shared__ float rowmin[128];

    const int tid  = threadIdx.x;
    const int w    = tid >> 5;        // wave id 0..7
    const int lane = tid & 31;
    const int r    = lane & 15;       // row within wave tile / B column
    const int h    = lane >> 4;       // lane half

    const int row0 = blockIdx.x * 128 + w * 16;       // first global row of wave
    const int b    = row0 >> 10;                      // batch (tile never crosses b)
    const int hw0  = row0 & (HW - 1);
    const float* zb = z + (size_t)b * C_DIM * HW + hw0 + r;

    // staging mapping: thread -> (code column sn, dim base cb)
    const int sn = tid >> 4;              // 0..15 : code within tile (coalesced)
    const int cb = (tid & 15) * 16;       // dim base, 16 consecutive dims
    const unsigned long long emb_u64 = (unsigned long long)(uintptr_t)emb;
    const uint32_t eb_base = (uint32_t)(uintptr_t)&ebuf[0][0];  // LDS byte offset

    // kick off tile 0 DMA before doing anything else
    stage_tile_async(eb_base, emb_u64, 0, sn, cb);

    // ---- load A panel into registers (WMMA A layout: K = 2h + j) ----
    v2f a[64];
#pragma unroll
    for (int f = 0; f < 64; ++f) {
        const int c = 4 * f + 2 * h;
        a[f].x = zb[(size_t)c * HW];
        a[f].y = zb[(size_t)(c + 1) * HW];
    }

    // ---- per-row ||z||^2 via in-register partials + shuffles ----
    float znp = 0.f;
#pragma unroll
    for (int f = 0; f < 64; ++f) znp += a[f].x * a[f].x + a[f].y * a[f].y;
    znp += __shfl_xor(znp, 16, 32);       // combine lane halves -> full row norm
    float zn[8];
#pragma unroll
    for (int g = 0; g < 8; ++g) zn[g] = __shfl(znp, g + 8 * h, 32);  // norm of D-row g+8h

    float vmin[8];
    int   vidx[8];
#pragma unroll
    for (int g = 0; g < 8; ++g) { vmin[g] = 3.4e38f; vidx[g] = 0; }

    // tile 0 visible to all waves
    wait_async_zero();
    __syncthreads();

    for (int step = 0; step < 64; ++step) {
        const int k0 = step * 16;
        // prefetch next tile into the other buffer; safe: that buffer's last
        // readers finished before the barrier that ended step-1.
        if (step < 63)
            stage_tile_async(eb_base + (uint32_t)(((step + 1) & 1) * 16384),
                             emb_u64, k0 + 16, sn, cb);

        // ---- 64-deep fp32 WMMA accumulation over the 256-dim axis ----
        const float* ebp = &ebuf[step & 1][r * 4 + 2 * h];
        v8f acc = {};
#pragma unroll
        for (int f = 0; f < 64; ++f) {
            v2f bb = *(const v2f*)(ebp + f * 64);
            acc = __builtin_amdgcn_wmma_f32_16x16x4_f32(
                false, a[f], false, bb, (short)0, acc, false, false);
        }

        // ---- distances + running argmin (lane column = code k0+r) ----
        const float en = enorm[k0 + r];
#pragma unroll
        for (int g = 0; g < 8; ++g) {
            const float d = zn[g] + en - 2.0f * acc[g];
            if (d < vmin[g]) { vmin[g] = d; vidx[g] = k0 + r; }
        }

        // next tile's DMA done + visible to all waves
        wait_async_zero();
        __syncthreads();
    }

    // ---- cross-lane argmin over the 16 columns in each half ----
#pragma unroll
    for (int g = 0; g < 8; ++g) {
#pragma unroll
        for (int off = 1; off < 16; off <<= 1) {
            float om = __shfl_xor(vmin[g], off, 32);
            int   oi = __shfl_xor(vidx[g], off, 32);
            if (om < vmin[g] || (om == vmin[g] && oi < vidx[g])) {
                vmin[g] = om; vidx[g] = oi;
            }
        }
    }

    if (r == 0) {   // lanes 0 and 16: write 8 rows each
#pragma unroll
        for (int g = 0; g < 8; ++g) {
            const int rr = g + 8 * h;
            const int n  = row0 + rr;
            const float md = vmin[g];
            const int   id = vidx[g];
            scores_out[n] = expf(-md * 0.1f);
            idxf_out[n]   = (float)id;
            idx_ws[n]     = id;
            atomicAdd(&counts[id], 1);
            rowmin[w * 16 + rr] = md;
        }
    }
    __syncthreads();
    if (tid == 0) {
        float s = 0.f;
        for (int i = 0; i < 128; ++i) s += rowmin[i];
        partials[blockIdx.x] = s;
    }
}

// ---------------------------------------------------------------------------
// Kernel 4: z_q gather, written directly in NCHW (coalesced along hw)
// ---------------------------------------------------------------------------
__global__ __launch_bounds__(256) void vq_out(const float* __restrict__ emb,
                                              const int* __restrict__ idx_ws,
                                              float* __restrict__ out) {
    const size_t t = (size_t)blockIdx.x * 256 + threadIdx.x;   // < 8388608
    const int hw = (int)(t & (HW - 1));
    const int c  = (int)((t >> 10) & (C_DIM - 1));
    const int b  = (int)(t >> 18);
    const int n  = b * HW + hw;
    out[t] = emb[(size_t)idx_ws[n] * C_DIM + c];
}

// ---------------------------------------------------------------------------
// Kernel 5: deterministic single-block finalize (loss, perplexity, mean_dist)
// ---------------------------------------------------------------------------
__global__ __launch_bounds__(256) void vq_finalize(const float* __restrict__ enorm,
                                                   const float* __restrict__ Ec,
                                                   const float* __restrict__ Zc,
                                                   const float* __restrict__ SqC,
                                                   const float* __restrict__ partials,
                                                   const int* __restrict__ counts,
                                                   float* __restrict__ scalars) {
    __shared__ float red[256];
    __shared__ float res[5];
    const int tid = threadIdx.x;

#define BLOCK_REDUCE(slot)                                                    \
    red[tid] = v; __syncthreads();                                            \
    for (int off = 128; off > 0; off >>= 1) {                                 \
        if (tid < off) red[tid] += red[tid + off];                            \
        __syncthreads();                                                      \
    }                                                                         \
    if (tid == 0) res[slot] = red[0];                                         \
    __syncthreads();

    float v;
    v = partials[tid];                                   BLOCK_REDUCE(0)  // sum min_d
    v = 0.f; for (int i = tid; i < K_CB; i += 256) v += enorm[i];
                                                         BLOCK_REDUCE(1)  // sum ||e||^2
    v = SqC[tid];                                        BLOCK_REDUCE(2)  // sum ||z||^2
    v = Zc[tid] * Ec[tid];                               BLOCK_REDUCE(3)  // Z . E
    v = 0.f;
    for (int i = tid; i < K_CB; i += 256) {
        const float em = (float)counts[i] * (1.0f / (float)N_TOT);
        v += em * logf(em + 1e-10f);
    }
                                                         BLOCK_REDUCE(4)  // -entropy
#undef BLOCK_REDUCE

    if (tid == 0) {
        const float sum_min = res[0];
        const float sum_en  = res[1];
        const float sum_sq  = res[2];
        const float sum_ze  = res[3];
        scalars[0] = 1.25f * sum_min / ((float)N_TOT * (float)C_DIM);        // loss
        scalars[1] = expf(-res[4]);                                          // perplexity
        scalars[2] = ((float)K_CB * sum_sq + (float)N_TOT * sum_en - 2.f * sum_ze)
                     / ((float)N_TOT * (float)K_CB);                         // mean_distance
    }
}

// ---------------------------------------------------------------------------
extern "C" void kernel_launch(void* const* d_in, const int* in_sizes, int n_in,
                              void* d_out, int out_size, void* d_ws, size_t ws_size,
                              hipStream_t stream) {
    const float* z   = (const float*)d_in[0];
    const float* emb = (const float*)d_in[1];
    float* out = (float*)d_out;

    float* w        = (float*)d_ws;
    float* enorm    = w;                 // 1024
    float* Ec       = w + 1024;          // 256
    float* Zc       = w + 1280;          // 256
    float* SqC      = w + 1536;          // 256
    float* partials = w + 1792;          // 256
    int*   counts   = (int*)(w + 2048);  // 1024
    int*   idx_ws   = (int*)(w + 3072);  // 32768

    float* out_main = out;               // [0, 8388608)
    float* scalars  = out + BCHW;        // loss, perplexity, mean_distance
    float* scores   = out + BCHW + 3;    // 32768
    float* idxf     = out + BCHW + 3 + N_TOT; // 32768

    hipMemsetAsync(counts, 0, K_CB * sizeof(int), stream);
    vq_prep_emb<<<(K_CB + C_DIM + 255) / 256, 256, 0, stream>>>(emb, enorm, Ec);
    vq_prep_z  <<<C_DIM, 256, 0, stream>>>(z, Zc, SqC);
    vq_main    <<<N_TOT / 128, 256, 0, stream>>>(z, emb, enorm, scores, idxf,
                                                 idx_ws, counts, partials);
    vq_out     <<<BCHW / 256, 256, 0, stream>>>(emb, idx_ws, out_main);
    vq_finalize<<<1, 256, 0, stream>>>(enorm, Ec, Zc, SqC, partials, counts, scalars);
}